// GNNRegressor_70454643523892
// MI455X (gfx1250) — compile-verified
//
#include <hip/hip_runtime.h>
#include <hip/hip_bf16.h>

#define NNODES 50000
#define NEDGES 800000
#define NGRAPH 128
#define FNODE  128
#define FEDGE  16
#define HDIM   256
#define GFDIM  32

typedef __attribute__((ext_vector_type(2))) float v2f;
typedef __attribute__((ext_vector_type(8))) float v8f;

__device__ __forceinline__ void atomAddF(float* p, float v) {
    unsafeAtomicAdd(p, v);
}

// Async global->LDS copy of 16 bytes (CDNA5 GLOBAL_LOAD_ASYNC_TO_LDS_B128,
// tracked by ASYNCcnt). ldsPtr's low 32 bits are the wave-relative LDS offset
// (generic-pointer aperture truncation rule, ISA 10.2).
__device__ __forceinline__ void asyncCopyB128(const float* gptr, float* ldsPtr) {
    unsigned ldsOff = (unsigned)(size_t)ldsPtr;
    unsigned long long ga = (unsigned long long)(size_t)gptr;
    asm volatile("global_load_async_to_lds_b128 %0, %1, off"
                 :: "v"(ldsOff), "v"(ga) : "memory");
}

__device__ __forceinline__ void waitAsync0() {
    asm volatile("s_wait_asynccnt 0" ::: "memory");
}

// ---------------------------------------------------------------------------
// Zero fill
// ---------------------------------------------------------------------------
__global__ __launch_bounds__(256) void k_zero(float* __restrict__ p, long n) {
    long i = (long)blockIdx.x * 256 + threadIdx.x;
    if (i < n) p[i] = 0.0f;
}

// ---------------------------------------------------------------------------
// Edge scatter: agg[dst] += relu(h[src] + edge_attr @ ew + eb)
// One wave (32 lanes) per edge; ew (16 x FIN) staged in LDS.
// ---------------------------------------------------------------------------
template <int FIN>
__global__ __launch_bounds__(256) void k_scatter(
    const float* __restrict__ h, const int* __restrict__ src,
    const int* __restrict__ dst, const float* __restrict__ ea,
    const float* __restrict__ ew, const float* __restrict__ eb,
    float* __restrict__ agg, int nedges)
{
    __shared__ float ews[16 * FIN];
    __shared__ float ebs[FIN];
    for (int i = threadIdx.x; i < 16 * FIN; i += 256) ews[i] = ew[i];
    for (int i = threadIdx.x; i < FIN; i += 256) ebs[i] = eb[i];
    __syncthreads();

    const int lane = threadIdx.x & 31;
    const int wave = threadIdx.x >> 5;
    const int e = blockIdx.x * 8 + wave;
    if (e >= nedges) return;

    const int s = src[e];
    const int d = dst[e];

    float eav[16];
#pragma unroll
    for (int j = 0; j < 16; ++j) eav[j] = ea[(long)e * 16 + j];

    const float* hrow = h + (long)s * FIN;
    float* arow = agg + (long)d * FIN;

    constexpr int CPL = FIN / 32;  // columns per lane
#pragma unroll
    for (int cc = 0; cc < CPL; ++cc) {
        const int c = cc * 32 + lane;
        float et = ebs[c];
#pragma unroll
        for (int j = 0; j < 16; ++j) et = fmaf(eav[j], ews[j * FIN + c], et);
        float m = hrow[c] + et;
        m = fmaxf(m, 0.0f);
        atomAddF(&arow[c], m);
    }
}

// ---------------------------------------------------------------------------
// WMMA fp32 GEMM: C[nrows x 256] = act( (A (+A2)) @ W + bias )
//   A: nrows x K row-major, W: K x 256 row-major.
//   Block = 256 threads = 8 waves; block tile 128x64; wave tile 32x32
//   (2x2 accumulators of v_wmma_f32_16x16x4_f32).
//   K staged in 16-wide slices via async global->LDS copies, double-buffered.
// ---------------------------------------------------------------------------
template <int K, bool RELU, bool HAS_A2>
__global__ __launch_bounds__(256) void k_gemm(
    const float* __restrict__ A, const float* __restrict__ A2,
    const float* __restrict__ W, const float* __restrict__ bias,
    float* __restrict__ C, int nrows)
{
    __shared__ float As[2][128][17];                  // 128 rows x 16 K-slice (+1 pad)
    __shared__ float A2s[HAS_A2 ? 2 : 1][128][17];
    __shared__ float Ws[2][16][64];                   // 16 K-rows x 64 cols

    const int tid   = threadIdx.x;
    const int lane  = tid & 31;
    const int wave  = tid >> 5;
    const int wr    = wave >> 1;       // 0..3 : wave row in block
    const int wc    = wave & 1;        // 0..1 : wave col in block
    const int l16   = lane & 15;
    const int lhalf = lane >> 4;       // 0/1

    const int blockRow = blockIdx.x * 128;
    const int blockCol = blockIdx.y * 64;

    // per-thread staging coordinates (invariant across K-slices)
    int rA[2], c4A[2], gRowA[2];
#pragma unroll
    for (int i = 0; i < 2; ++i) {
        const int idx = tid + i * 256;     // 0..511
        rA[i]  = idx >> 2;
        c4A[i] = (idx & 3) * 4;
        int grow = blockRow + rA[i];
        if (grow >= nrows) grow = nrows - 1;   // clamp: garbage rows never stored
        gRowA[i] = grow;
    }
    const int rW  = tid >> 4;           // 0..15
    const int c4W = (tid & 15) * 4;

    auto issueSlice = [&](int k0, int buf) {
#pragma unroll
        for (int i = 0; i < 2; ++i) {
            asyncCopyB128(A + (long)gRowA[i] * K + k0 + c4A[i], &As[buf][rA[i]][c4A[i]]);
            if constexpr (HAS_A2)
                asyncCopyB128(A2 + (long)gRowA[i] * K + k0 + c4A[i], &A2s[buf][rA[i]][c4A[i]]);
        }
        asyncCopyB128(W + (long)(k0 + rW) * HDIM + blockCol + c4W, &Ws[buf][rW][c4W]);
    };

    v8f acc[2][2] = {};

    issueSlice(0, 0);
    waitAsync0();
    __syncthreads();

    constexpr int NS = K / 16;
    for (int s = 0; s < NS; ++s) {
        const int cur = s & 1;
        if (s + 1 < NS) issueSlice((s + 1) * 16, cur ^ 1);

#pragma unroll
        for (int kk = 0; kk < 4; ++kk) {
            // A operand (16x4 f32): VGPR0 = {K=k, K=k+2}, VGPR1 = {K=k+1, K=k+3}
            v2f a[2], b[2];
#pragma unroll
            for (int t = 0; t < 2; ++t) {
                const int row = wr * 32 + t * 16 + l16;
                const int ka  = kk * 4 + lhalf * 2;
                float ax = As[cur][row][ka];
                float ay = As[cur][row][ka + 1];
                if constexpr (HAS_A2) {
                    ax += A2s[cur][row][ka];
                    ay += A2s[cur][row][ka + 1];
                }
                a[t].x = ax;
                a[t].y = ay;
                const int col = wc * 32 + t * 16 + l16;
                b[t].x = Ws[cur][ka][col];
                b[t].y = Ws[cur][ka + 1][col];
            }
#pragma unroll
            for (int tr = 0; tr < 2; ++tr)
#pragma unroll
                for (int tc = 0; tc < 2; ++tc)
                    acc[tr][tc] = __builtin_amdgcn_wmma_f32_16x16x4_f32(
                        false, a[tr], false, b[tc],
                        (short)0, acc[tr][tc], false, false);
        }

        waitAsync0();          // next buffer filled
        __syncthreads();       // everyone done reading cur buffer
    }

    // epilogue: C VGPR p holds row (p + 8*lhalf) of the 16x16 tile, col = l16
#pragma unroll
    for (int tr = 0; tr < 2; ++tr) {
#pragma unroll
        for (int tc = 0; tc < 2; ++tc) {
            const int col = blockCol + wc * 32 + tc * 16 + l16;
            const float bv = bias[col];
#pragma unroll
            for (int p = 0; p < 8; ++p) {
                const int row = blockRow + wr * 32 + tr * 16 + p + lhalf * 8;
                if (row < nrows) {
                    float v = acc[tr][tc][p] + bv;
                    if (RELU) v = fmaxf(v, 0.0f);
                    C[(long)row * HDIM + col] = v;
                }
            }
        }
    }
}

// ---------------------------------------------------------------------------
// BatchNorm statistics: per-column sum and sum-of-squares over all rows.
// ---------------------------------------------------------------------------
__global__ __launch_bounds__(256) void k_bnstats(
    const float* __restrict__ h, float* __restrict__ stats, int nrows)
{
    const int c  = threadIdx.x;          // column 0..255
    const int r0 = blockIdx.x * 512;
    int r1 = r0 + 512; if (r1 > nrows) r1 = nrows;
    float s = 0.f, s2 = 0.f;
    for (int r = r0; r < r1; ++r) {
        float v = h[(long)r * HDIM + c];
        s += v; s2 += v * v;
    }
    atomAddF(&stats[c], s);
    atomAddF(&stats[HDIM + c], s2);
}

__global__ __launch_bounds__(256) void k_bnapply(
    float* __restrict__ h, const float* __restrict__ stats,
    const float* __restrict__ gamma, const float* __restrict__ beta, int nrows)
{
    const long idx = (long)blockIdx.x * 256 + threadIdx.x;
    if (idx >= (long)nrows * HDIM) return;
    const int c = (int)(idx & (HDIM - 1));
    const float invN = 1.0f / (float)nrows;
    const float mu  = stats[c] * invN;
    const float var = stats[HDIM + c] * invN - mu * mu;
    const float inv = rsqrtf(var + 1e-5f);
    const float v = gamma[c] * (h[idx] - mu) * inv + beta[c];
    h[idx] = fmaxf(v, 0.0f);
}

// ---------------------------------------------------------------------------
// Graph pooling: g[batch[n]] += h[n]
// ---------------------------------------------------------------------------
__global__ __launch_bounds__(256) void k_pool(
    const float* __restrict__ h, const int* __restrict__ batch,
    float* __restrict__ g, int nrows)
{
    const long idx = (long)blockIdx.x * 256 + threadIdx.x;
    if (idx >= (long)nrows * HDIM) return;
    const int n = (int)(idx >> 8);
    const int c = (int)(idx & (HDIM - 1));
    atomAddF(&g[(long)batch[n] * HDIM + c], h[idx]);
}

// ---------------------------------------------------------------------------
// Small dense heads (M <= 128) - scalar dot products
// ---------------------------------------------------------------------------
__global__ __launch_bounds__(256) void k_dense(
    const float* __restrict__ X, const float* __restrict__ W,
    const float* __restrict__ b, float* __restrict__ Y,
    int M, int Kd, int Nd, int relu)
{
    const int idx = blockIdx.x * 256 + threadIdx.x;
    if (idx >= M * Nd) return;
    const int i = idx / Nd, j = idx % Nd;
    float s = b[j];
    for (int k = 0; k < Kd; ++k) s = fmaf(X[i * Kd + k], W[k * Nd + j], s);
    if (relu) s = fmaxf(s, 0.0f);
    Y[idx] = s;
}

__global__ __launch_bounds__(128) void k_concat(
    const float* __restrict__ g, const float* __restrict__ gl,
    float* __restrict__ gcat)
{
    const int idx = blockIdx.x * 128 + threadIdx.x;   // 128*288
    if (idx >= NGRAPH * (HDIM + GFDIM)) return;
    const int i = idx / (HDIM + GFDIM), j = idx % (HDIM + GFDIM);
    gcat[idx] = (j < HDIM) ? g[i * HDIM + j] : gl[i * GFDIM + (j - HDIM)];
}

__global__ __launch_bounds__(128) void k_rownorm(float* __restrict__ Z, int M, int Nd) {
    const int i = blockIdx.x * 128 + threadIdx.x;
    if (i >= M) return;
    float s = 0.f;
    for (int j = 0; j < Nd; ++j) { float v = Z[i * Nd + j]; s += v * v; }
    const float inv = 1.0f / fmaxf(sqrtf(s), 1e-12f);
    for (int j = 0; j < Nd; ++j) Z[i * Nd + j] *= inv;
}

__global__ __launch_bounds__(256) void k_edl(
    const float* __restrict__ gf2, const float* __restrict__ W,
    const float* __restrict__ b, float* __restrict__ out, int M)
{
    const int idx = blockIdx.x * 256 + threadIdx.x;
    if (idx >= M * 4) return;
    const int i = idx / 4, j = idx % 4;
    float s = b[j];
    for (int k = 0; k < 128; ++k) s = fmaf(gf2[i * 128 + k], W[k * 4 + j], s);
    float r;
    if (j == 0) {
        r = s;
    } else {
        // stable softplus: log1p(exp(-|x|)) + max(x, 0)
        float sp = log1pf(expf(-fabsf(s))) + fmaxf(s, 0.0f);
        r = (j == 2) ? (sp + 1.0f + 1e-6f) : (sp + 1e-6f);
    }
    out[idx] = r;
}

// ---------------------------------------------------------------------------
// Orchestration
// ---------------------------------------------------------------------------
extern "C" void kernel_launch(void* const* d_in, const int* in_sizes, int n_in,
                              void* d_out, int out_size, void* d_ws, size_t ws_size,
                              hipStream_t stream)
{
    const float* x      = (const float*)d_in[0];
    const int*   ei     = (const int*)  d_in[1];
    const float* ea     = (const float*)d_in[2];
    const int*   batch  = (const int*)  d_in[3];
    const float* gfeat  = (const float*)d_in[4];
    const float* c0_w1  = (const float*)d_in[5];
    const float* c0_b1  = (const float*)d_in[6];
    const float* c0_w2  = (const float*)d_in[7];
    const float* c0_b2  = (const float*)d_in[8];
    const float* c0_ew  = (const float*)d_in[9];
    const float* c0_eb  = (const float*)d_in[10];
    const float* cs_w1  = (const float*)d_in[11];
    const float* cs_b1  = (const float*)d_in[12];
    const float* cs_w2  = (const float*)d_in[13];
    const float* cs_b2  = (const float*)d_in[14];
    const float* cs_ew  = (const float*)d_in[15];
    const float* cs_eb  = (const float*)d_in[16];
    const float* bn_g   = (const float*)d_in[17];
    const float* bn_b   = (const float*)d_in[18];
    const float* pw1    = (const float*)d_in[19];
    const float* pb1    = (const float*)d_in[20];
    const float* pw2    = (const float*)d_in[21];
    const float* pb2    = (const float*)d_in[22];
    const float* fw1    = (const float*)d_in[23];
    const float* fb1    = (const float*)d_in[24];
    const float* fw2    = (const float*)d_in[25];
    const float* fb2    = (const float*)d_in[26];
    const float* edl_w  = (const float*)d_in[27];
    const float* edl_b  = (const float*)d_in[28];

    const int* srcI = ei;
    const int* dstI = ei + NEDGES;

    float* ws    = (float*)d_ws;
    float* agg   = ws;                                  // N x 256
    float* tbuf  = ws + (long)NNODES * HDIM;            // N x 256
    float* hbuf  = ws + 2L * NNODES * HDIM;             // N x 256
    float* stats = ws + 3L * NNODES * HDIM;             // 512
    float* z1    = stats + 512;                         // 128 x 128
    float* gcat  = z1 + NGRAPH * 128;                   // 128 x 288
    float* gf1   = gcat + NGRAPH * (HDIM + GFDIM);      // 128 x 256
    float* gf2   = gf1 + NGRAPH * HDIM;                 // 128 x 128

    float* outF = (float*)d_out;
    float* edl  = outF;                                 // 128 x 4
    float* z    = outF + NGRAPH * 4;                    // 128 x 64
    float* g    = outF + NGRAPH * 4 + NGRAPH * 64;      // 128 x 256

    const dim3 blk(256);
    const dim3 gemmGrid((NNODES + 127) / 128, HDIM / 64);
    const int scatBlocks = (NEDGES + 7) / 8;
    const int elemBlocks = (int)(((long)NNODES * HDIM + 255) / 256);

    // ---- layer 0 (K = 128) ----
    k_zero<<<(int)(((long)NNODES * FNODE + 255) / 256), blk, 0, stream>>>(agg, (long)NNODES * FNODE);
    k_scatter<FNODE><<<scatBlocks, blk, 0, stream>>>(x, srcI, dstI, ea, c0_ew, c0_eb, agg, NEDGES);
    k_gemm<FNODE, true, true><<<gemmGrid, blk, 0, stream>>>(x, agg, c0_w1, c0_b1, tbuf, NNODES);
    k_gemm<HDIM, false, false><<<gemmGrid, blk, 0, stream>>>(tbuf, nullptr, c0_w2, c0_b2, hbuf, NNODES);
    k_zero<<<2, blk, 0, stream>>>(stats, 512);
    k_bnstats<<<(NNODES + 511) / 512, blk, 0, stream>>>(hbuf, stats, NNODES);
    k_bnapply<<<elemBlocks, blk, 0, stream>>>(hbuf, stats, bn_g, bn_b, NNODES);

    // ---- layers 1..3 (K = 256) ----
    for (int l = 0; l < 3; ++l) {
        k_zero<<<elemBlocks, blk, 0, stream>>>(agg, (long)NNODES * HDIM);
        k_scatter<HDIM><<<scatBlocks, blk, 0, stream>>>(
            hbuf, srcI, dstI, ea, cs_ew + (long)l * 16 * HDIM, cs_eb + (long)l * HDIM, agg, NEDGES);
        k_gemm<HDIM, true, true><<<gemmGrid, blk, 0, stream>>>(
            hbuf, agg, cs_w1 + (long)l * HDIM * HDIM, cs_b1 + (long)l * HDIM, tbuf, NNODES);
        k_gemm<HDIM, false, false><<<gemmGrid, blk, 0, stream>>>(
            tbuf, nullptr, cs_w2 + (long)l * HDIM * HDIM, cs_b2 + (long)l * HDIM, hbuf, NNODES);
        k_zero<<<2, blk, 0, stream>>>(stats, 512);
        k_bnstats<<<(NNODES + 511) / 512, blk, 0, stream>>>(hbuf, stats, NNODES);
        k_bnapply<<<elemBlocks, blk, 0, stream>>>(
            hbuf, stats, bn_g + (long)(l + 1) * HDIM, bn_b + (long)(l + 1) * HDIM, NNODES);
    }

    // ---- pooling + heads ----
    k_zero<<<(out_size + 255) / 256, blk, 0, stream>>>(outF, out_size);
    k_pool<<<elemBlocks, blk, 0, stream>>>(hbuf, batch, g, NNODES);

    k_dense<<<(NGRAPH * 128 + 255) / 256, blk, 0, stream>>>(g, pw1, pb1, z1, NGRAPH, HDIM, 128, 1);
    k_dense<<<(NGRAPH * 64 + 255) / 256, blk, 0, stream>>>(z1, pw2, pb2, z, NGRAPH, 128, 64, 0);
    k_rownorm<<<1, 128, 0, stream>>>(z, NGRAPH, 64);

    k_concat<<<(NGRAPH * (HDIM + GFDIM) + 127) / 128, 128, 0, stream>>>(g, gfeat, gcat);
    k_dense<<<(NGRAPH * HDIM + 255) / 256, blk, 0, stream>>>(gcat, fw1, fb1, gf1, NGRAPH, HDIM + GFDIM, HDIM, 1);
    k_dense<<<(NGRAPH * 128 + 255) / 256, blk, 0, stream>>>(gf1, fw2, fb2, gf2, NGRAPH, HDIM, 128, 1);
    k_edl<<<(NGRAPH * 4 + 255) / 256, blk, 0, stream>>>(gf2, edl_w, edl_b, edl, NGRAPH);
}